// BatchedGCN_90280212562329
// MI455X (gfx1250) — compile-verified
//
#include <hip/hip_runtime.h>
#include <hip/hip_bf16.h>

typedef __attribute__((ext_vector_type(2))) float v2f;
typedef __attribute__((ext_vector_type(4))) float v4f;
typedef __attribute__((ext_vector_type(8))) float v8f;

#define GCN_N   4096
#define GCN_E   64
#define GCN_B   8
#define EPS_F   1e-8f

// ---------------------------------------------------------------------------
// Kernel 1: H = X @ W + b    ([B*N, 64] = [B*N, 64] @ [64, 64] + [64])
// Tiny (268 MFLOP total): W staged in LDS, 4 threads per row, 16 cols each.
// ---------------------------------------------------------------------------
__global__ void __launch_bounds__(256)
linear_kernel(const float* __restrict__ X, const float* __restrict__ W,
              const float* __restrict__ bias, float* __restrict__ H)
{
    __shared__ float Wl[64 * 64];
    __shared__ float bl[64];
    const int t = threadIdx.x;
#pragma unroll
    for (int i = 0; i < 16; ++i) Wl[t + i * 256] = W[t + i * 256];
    if (t < 64) bl[t] = bias[t];
    __syncthreads();

    const int row = blockIdx.x * 64 + (t >> 2);   // global row in [0, B*N)
    const int c0  = (t & 3) * 16;
    const float* Xr = X + (size_t)row * GCN_E;

    float acc[16];
#pragma unroll
    for (int c = 0; c < 16; ++c) acc[c] = bl[c0 + c];
    for (int k = 0; k < 64; ++k) {
        const float xv = Xr[k];
#pragma unroll
        for (int c = 0; c < 16; ++c)
            acc[c] = fmaf(xv, Wl[k * 64 + c0 + c], acc[c]);
    }
    float* Hr = H + (size_t)row * GCN_E;
#pragma unroll
    for (int c = 0; c < 16; ++c) Hr[c0 + c] = acc[c];
}

// ---------------------------------------------------------------------------
// Kernel 2 (per batch): dinv[n] = rsqrt(max(sum_m A[n,m] + 1, eps))
// One wave32 per row; regular-temporal loads intentionally warm L2 (64 MB of
// A per batch fits the 192 MB global L2) for the aggregation pass.
// ---------------------------------------------------------------------------
__global__ void __launch_bounds__(256)
degree_rsqrt_kernel(const float* __restrict__ A, float* __restrict__ dinv, int batch)
{
    const int lane = threadIdx.x & 31;
    const int wave = threadIdx.x >> 5;
    const int row  = blockIdx.x * 8 + wave;
    const float* Ar = A + ((size_t)batch * GCN_N + row) * GCN_N;

    float s = 0.0f;
    for (int i = lane; i < GCN_N / 4; i += 32) {
        v4f v = *(const v4f*)(Ar + (size_t)i * 4);
        s += v.x + v.y + v.z + v.w;
    }
#pragma unroll
    for (int off = 16; off > 0; off >>= 1)
        s += __shfl_xor(s, off, 32);

    if (lane == 0) {
        const float d = fmaxf(s + 1.0f, EPS_F);       // +1 from the identity
        dinv[batch * GCN_N + row] = rsqrtf(d);
    }
}

// ---------------------------------------------------------------------------
// Kernel 3 (per batch): out[n,:] += dinv[n] * (Ahat[n, kslice] @ (dinv[k]*H[k,:]))
// 256 threads = 8 waves; WG tile 64 rows x 64 cols; wave w -> rows (w>>1)*16,
// cols (w&1)*32 => two v8f accumulators. K split 2-way -> 128 WGs/batch with
// fp32 atomic accumulation into pre-zeroed out.
//
// LDS: At[r][k] row-major; HtT[e][k] TRANSPOSED so both A and B fragments are
// single contiguous 8-byte ds_load_b64 into even-aligned VGPR pairs (no
// repacking movs). PAD=68 floats: rows 16B-aligned, frag loads hit all 64
// banks exactly once (bank = 4*idx + k&3).
//
// Software pipeline: next chunk's A/H/dinv prefetched into registers while
// the WMMA loop consumes the current LDS tiles (single-buffered LDS, no
// extra barrier cost, hides HBM/L2 latency).
// ---------------------------------------------------------------------------
#define KC     64
#define PAD    68
#define ROWS   64
#define KSLICE 2048

__global__ void __launch_bounds__(256)
gcn_aggregate_kernel(const float* __restrict__ A, const float* __restrict__ H,
                     const float* __restrict__ dinv, float* __restrict__ out,
                     int batch)
{
    __shared__ float At [ROWS * PAD];   // At[r*PAD + k]
    __shared__ float HtT[GCN_E * PAD];  // HtT[e*PAD + k]  (transposed, scaled)

    const int t    = threadIdx.x;
    const int lane = t & 31;
    const int w    = t >> 5;
    const int grow = blockIdx.x * ROWS;            // output row tile base
    const int k0b  = blockIdx.y * KSLICE;          // K slice base

    const float* Ab = A + (size_t)batch * GCN_N * GCN_N + (size_t)grow * GCN_N;
    const float* Hb = H + (size_t)batch * GCN_N * GCN_E;
    const float* db = dinv + (size_t)batch * GCN_N;

    const int wr = (w >> 1) * 16;                  // wave row offset in tile
    const int wc = (w & 1) * 32;                   // wave col offset
    const int m  = lane & 15;                      // M (A frag) / N (B frag)
    const int kh = (lane >> 4) << 1;               // K sub-offset: 0 or 2

    // per-thread staging tasks: f4 = t + i*256 -> r = f4>>4, c4 = (f4&15)*4
    v4f  aReg[4], hReg[4];
    float dReg[4];

#pragma unroll
    for (int i = 0; i < 4; ++i) {                  // preload chunk 0
        const int f4 = t + i * 256;
        const int r  = f4 >> 4;
        const int c4 = (f4 & 15) << 2;
        aReg[i] = __builtin_nontemporal_load(
                      (const v4f*)(Ab + (size_t)r * GCN_N + k0b + c4));
        hReg[i] = *(const v4f*)(Hb + (size_t)(k0b + r) * GCN_E + c4);
        dReg[i] = db[k0b + r];
    }

    v8f acc0 = {}, acc1 = {};

    for (int k0 = k0b; k0 < k0b + KSLICE; k0 += KC) {
        // ---- commit staged registers to LDS (H scaled by dinv[k], transposed)
#pragma unroll
        for (int i = 0; i < 4; ++i) {
            const int f4 = t + i * 256;
            const int r  = f4 >> 4;
            const int c4 = (f4 & 15) << 2;
            *(v4f*)&At[r * PAD + c4] = aReg[i];
            const float s = dReg[i];
            HtT[(c4 + 0) * PAD + r] = hReg[i].x * s;
            HtT[(c4 + 1) * PAD + r] = hReg[i].y * s;
            HtT[(c4 + 2) * PAD + r] = hReg[i].z * s;
            HtT[(c4 + 3) * PAD + r] = hReg[i].w * s;
        }
        __syncthreads();

        // ---- prefetch next chunk into registers (overlaps with WMMA loop)
        if (k0 + KC < k0b + KSLICE) {
            const int kn = k0 + KC;
#pragma unroll
            for (int i = 0; i < 4; ++i) {
                const int f4 = t + i * 256;
                const int r  = f4 >> 4;
                const int c4 = (f4 & 15) << 2;
                aReg[i] = __builtin_nontemporal_load(
                              (const v4f*)(Ab + (size_t)r * GCN_N + kn + c4));
                hReg[i] = *(const v4f*)(Hb + (size_t)(kn + r) * GCN_E + c4);
                dReg[i] = db[kn + r];
            }
        }

        // ---- WMMA loop: all fragments are single b64 LDS loads
#pragma unroll
        for (int kk = 0; kk < KC; kk += 4) {
            v2f a  = *(const v2f*)&At [(wr + m)      * PAD + kk + kh];
            v2f b0 = *(const v2f*)&HtT[(wc + m)      * PAD + kk + kh];
            v2f b1 = *(const v2f*)&HtT[(wc + 16 + m) * PAD + kk + kh];
            acc0 = __builtin_amdgcn_wmma_f32_16x16x4_f32(
                       false, a, false, b0, (short)0, acc0, false, false);
            acc1 = __builtin_amdgcn_wmma_f32_16x16x4_f32(
                       false, a, false, b1, (short)0, acc1, false, false);
        }
        __syncthreads();
    }

    // ---- epilogue: row normalization + atomic accumulation over K slices
    const int rhalf = (lane >> 4) << 3;            // 0 or 8 (C/D layout)
#pragma unroll
    for (int r = 0; r < 8; ++r) {
        const int row = grow + wr + r + rhalf;
        const float s = db[row];
        float* o = out + ((size_t)batch * GCN_N + row) * GCN_E + wc + m;
        unsafeAtomicAdd(o,      acc0[r] * s);
        unsafeAtomicAdd(o + 16, acc1[r] * s);
    }
}

// ---------------------------------------------------------------------------
// Launch: H once, then per batch (degree -> aggregate) so each batch's 64 MB
// slice of A stays resident in the 192 MB L2 between the two passes.
// ---------------------------------------------------------------------------
extern "C" void kernel_launch(void* const* d_in, const int* in_sizes, int n_in,
                              void* d_out, int out_size, void* d_ws, size_t ws_size,
                              hipStream_t stream)
{
    (void)n_in; (void)ws_size; (void)in_sizes;
    const float* X    = (const float*)d_in[0];   // [B,N,64]
    const float* A    = (const float*)d_in[1];   // [B,N,N]
    const float* W    = (const float*)d_in[2];   // [64,64]
    const float* bias = (const float*)d_in[3];   // [64]
    float* out = (float*)d_out;

    float* dinv = (float*)d_ws;                               // B*N floats
    float* Hws  = (float*)((char*)d_ws + (size_t)GCN_B * GCN_N * sizeof(float));

    hipMemsetAsync(d_out, 0, (size_t)out_size * sizeof(float), stream);

    linear_kernel<<<(GCN_B * GCN_N) / 64, 256, 0, stream>>>(X, W, bias, Hws);

    for (int b = 0; b < GCN_B; ++b) {
        degree_rsqrt_kernel<<<GCN_N / 8, 256, 0, stream>>>(A, dinv, b);
        gcn_aggregate_kernel<<<dim3(GCN_N / ROWS, 2), 256, 0, stream>>>(
            A, Hws, dinv, out, b);
    }
}